// ProbHypernet_52432960750125
// MI455X (gfx1250) — compile-verified
//
#include <hip/hip_runtime.h>
#include <math.h>

#define MDIM 256
#define INDIM 511
#define HDIM 256
#define RDIM 512
#define ODIM 512

typedef __attribute__((ext_vector_type(2))) float v2f;
typedef __attribute__((ext_vector_type(8))) float v8f;

// ---------------------------------------------------------------------------
// f32 WMMA (V_WMMA_F32_16X16X4_F32, wave32, one wave per 16x16 tile).
// A 16x4 frag: lanes 0-15 hold (K=k0,k0+1) of row M=lane; lanes 16-31 hold
// (K=k0+2,k0+3) of row M=lane-16.  B 4x16 frag mirrors it (N=lane%16).
// C/D: VGPR j holds row j (lanes 0-15) and row j+8 (lanes 16-31), N=lane%16.
// ---------------------------------------------------------------------------

// Zero-pad K=511 -> 512: xp[256][512], Wp[512][256] so the WMMA loops run
// branch-free (the round-1 asm showed saveexec/branch spam per masked load).
__global__ __launch_bounds__(256)
void k_pad(const float* __restrict__ x, const float* __restrict__ Wxh,
           float* __restrict__ xp, float* __restrict__ Wp)
{
    const int i = blockIdx.x * 256 + threadIdx.x;
    if (i < MDIM * 512) {
        const int m = i >> 9, k = i & 511;
        xp[i] = (k < INDIM) ? x[m * INDIM + k] : 0.0f;
    } else {
        const int j = i - MDIM * 512;          // linear index into [512,256]
        const int k = j >> 8;                  // row
        Wp[j] = (k < INDIM) ? Wxh[j] : 0.0f;   // Wxh is [511,256] row-major
    }
}

// h = relu(xp @ Wp + b_xh)   [256,512]@[512,256] -> [256,256]
__global__ __launch_bounds__(32)
void k_hyper_h(const float* __restrict__ xp, const float* __restrict__ Wp,
               const float* __restrict__ bias, float* __restrict__ h)
{
    const int lane = threadIdx.x;
    const int half = lane >> 4;
    const int lm   = lane & 15;
    const int mT = blockIdx.y, nT = blockIdx.x;
    const float* __restrict__ arow = xp + (mT * 16 + lm) * 512;
    const int col = nT * 16 + lm;
    v8f acc = {};
#pragma unroll 4
    for (int k = 0; k < 512; k += 4) {
        const int k0 = k + 2 * half;
        const v2f a = *(const v2f*)(arow + k0);
        v2f b;
        b.x = Wp[(k0    ) * HDIM + col];
        b.y = Wp[(k0 + 1) * HDIM + col];
        acc = __builtin_amdgcn_wmma_f32_16x16x4_f32(false, a, false, b,
                                                    (short)0, acc, false, false);
    }
    const float bv = bias[col];
#pragma unroll
    for (int j = 0; j < 8; ++j) {
        const int rr = mT * 16 + j + half * 8;
        const float v = acc[j] + bv;
        h[rr * HDIM + col] = v > 0.0f ? v : 0.0f;
    }
}

// Three hypernet heads from h [256,256]: z=0 -> mu, z=1 -> logvar_r (+sqrt var),
// z=2 -> logvar_c (+sqrt var).  K=256 exact, N=512.
__global__ __launch_bounds__(32)
void k_hyper_heads(const float* __restrict__ h,
                   const float* __restrict__ Wmu, const float* __restrict__ bmu,
                   const float* __restrict__ Wlr, const float* __restrict__ blr,
                   const float* __restrict__ Wlc, const float* __restrict__ blc,
                   float* __restrict__ mu,  float* __restrict__ sr,
                   float* __restrict__ lvr, float* __restrict__ sc,
                   float* __restrict__ lvc)
{
    const int lane = threadIdx.x;
    const int half = lane >> 4;
    const int lm   = lane & 15;
    const int mT = blockIdx.y, nT = blockIdx.x, z = blockIdx.z;
    const float* __restrict__ W  = (z == 0) ? Wmu : ((z == 1) ? Wlr : Wlc);
    const float* __restrict__ bi = (z == 0) ? bmu : ((z == 1) ? blr : blc);
    const float* __restrict__ arow = h + (mT * 16 + lm) * HDIM;
    const int col = nT * 16 + lm;
    v8f acc = {};
#pragma unroll 4
    for (int k = 0; k < HDIM; k += 4) {
        const int k0 = k + 2 * half;
        const v2f a = *(const v2f*)(arow + k0);
        v2f b;
        b.x = W[(k0    ) * RDIM + col];
        b.y = W[(k0 + 1) * RDIM + col];
        acc = __builtin_amdgcn_wmma_f32_16x16x4_f32(false, a, false, b,
                                                    (short)0, acc, false, false);
    }
    const float bv = bi[col];
#pragma unroll
    for (int j = 0; j < 8; ++j) {
        const int rr  = mT * 16 + j + half * 8;
        const float v = acc[j] + bv;
        const int idx = rr * RDIM + col;
        if (z == 0)      { mu[idx]  = v; }
        else if (z == 1) { lvr[idx] = v; sr[idx] = expf(0.5f * v); }
        else             { lvc[idx] = v; sc[idx] = expf(0.5f * v); }
    }
}

// outmu = avec @ M   [256,512]@[512,512] -> [256,512]
__global__ __launch_bounds__(32)
void k_mu_out(const float* __restrict__ avec, const float* __restrict__ Mw,
              float* __restrict__ outmu)
{
    const int lane = threadIdx.x;
    const int half = lane >> 4;
    const int lm   = lane & 15;
    const int mT = blockIdx.y, nT = blockIdx.x;
    const float* __restrict__ arow = avec + (mT * 16 + lm) * RDIM;
    const int col = nT * 16 + lm;
    v8f acc = {};
#pragma unroll 4
    for (int k = 0; k < RDIM; k += 4) {
        const int k0 = k + 2 * half;
        const v2f a = *(const v2f*)(arow + k0);
        v2f b;
        b.x = Mw[(k0    ) * ODIM + col];
        b.y = Mw[(k0 + 1) * ODIM + col];
        acc = __builtin_amdgcn_wmma_f32_16x16x4_f32(false, a, false, b,
                                                    (short)0, acc, false, false);
    }
#pragma unroll
    for (int j = 0; j < 8; ++j) {
        const int rr = mT * 16 + j + half * 8;
        outmu[rr * ODIM + col] = acc[j];
    }
}

// S_M[r] = sum_c M[r,c]^2
__global__ __launch_bounds__(256)
void k_msq(const float* __restrict__ Mw, float* __restrict__ SM)
{
    const int r = blockIdx.x * 256 + threadIdx.x;
    float s = 0.0f;
#pragma unroll 4
    for (int c = 0; c < ODIM; ++c) { const float v = Mw[r * ODIM + c]; s += v * v; }
    SM[r] = s;
}

// Per-sample: avec=xa*mu, bvec=xa*sqrt(var_r); KL partial sums.
__global__ __launch_bounds__(256)
void k_prep(const float* __restrict__ x,  const float* __restrict__ mu,
            const float* __restrict__ sr, const float* __restrict__ lvr,
            const float* __restrict__ sc, const float* __restrict__ lvc,
            const float* __restrict__ SM,
            float* __restrict__ avec, float* __restrict__ bvec,
            float* __restrict__ klp)
{
    __shared__ float sm[5 * 256];
    const int m = blockIdx.x, tid = threadIdx.x;
    float a_vr = 0.f, a_lvr = 0.f, a_m2 = 0.f, a_vc = 0.f, a_lvc = 0.f;
    for (int r = tid; r < RDIM; r += 256) {
        const float xa  = (r < INDIM) ? x[m * INDIM + r] : 1.0f;
        const float muv = mu[m * RDIM + r];
        const float srv = sr[m * RDIM + r];
        avec[m * RDIM + r] = xa * muv;
        bvec[m * RDIM + r] = xa * srv;
        a_vr  += srv * srv;
        a_lvr += lvr[m * RDIM + r];
        a_m2  += muv * muv * SM[r];
    }
    for (int c = tid; c < ODIM; c += 256) {
        const float scv = sc[m * ODIM + c];
        a_vc  += scv * scv;
        a_lvc += lvc[m * ODIM + c];
    }
    sm[0 * 256 + tid] = a_vr;  sm[1 * 256 + tid] = a_lvr;
    sm[2 * 256 + tid] = a_m2;  sm[3 * 256 + tid] = a_vc;
    sm[4 * 256 + tid] = a_lvc;
    __syncthreads();
    for (int s = 128; s > 0; s >>= 1) {
        if (tid < s) {
#pragma unroll
            for (int i = 0; i < 5; ++i) sm[i * 256 + tid] += sm[i * 256 + tid + s];
        }
        __syncthreads();
    }
    if (tid == 0) {
        klp[m * 8 + 0] = sm[0];       klp[m * 8 + 1] = sm[256];
        klp[m * 8 + 2] = sm[512];     klp[m * 8 + 3] = sm[768];
        klp[m * 8 + 4] = sm[1024];
    }
}

// Bandwidth-bound pass: stream all 268 MB of E once (non-temporal, b64/lane),
// out[m,c] = outmu[m,c] + sc[m,c] * sum_r bvec[m,r] * E[m,r,c].
// bvec index is block-uniform -> scalar loads; E loads coalesce across c.
__global__ __launch_bounds__(256)
void k_estream(const float* __restrict__ E, const float* __restrict__ bvec,
               const float* __restrict__ sc, const float* __restrict__ outmu,
               float* __restrict__ out)
{
    const int m  = blockIdx.x;
    const int c2 = threadIdx.x;                      // float2 column index
    const v2f* __restrict__ Eb = (const v2f*)(E + (size_t)m * RDIM * ODIM) + c2;
    const float* __restrict__ bb = bvec + m * RDIM;
    v2f acc = {0.0f, 0.0f};
#pragma unroll 8
    for (int r = 0; r < RDIM; ++r) {
        const v2f e = __builtin_nontemporal_load(Eb + (size_t)r * (ODIM / 2));
        const float bv = bb[r];
        acc.x += bv * e.x;
        acc.y += bv * e.y;
    }
    const int idx = m * ODIM + 2 * c2;
    out[idx    ] = outmu[idx    ] + sc[idx    ] * acc.x;
    out[idx + 1] = outmu[idx + 1] + sc[idx + 1] * acc.y;
}

// D_KL = mean_m 0.5*(Svr*Svc + Sm2 - r*c - c*Slvr - r*Slvc)
__global__ __launch_bounds__(256)
void k_kl(const float* __restrict__ klp, float* __restrict__ dkl)
{
    __shared__ float sm[256];
    const int tid = threadIdx.x;
    const float vr = klp[tid * 8 + 0], lr = klp[tid * 8 + 1], m2 = klp[tid * 8 + 2];
    const float vc = klp[tid * 8 + 3], lc = klp[tid * 8 + 4];
    const float rc = (float)(RDIM * ODIM);
    sm[tid] = 0.5f * (vr * vc + m2 - rc - (float)ODIM * lr - (float)RDIM * lc);
    __syncthreads();
    for (int s = 128; s > 0; s >>= 1) {
        if (tid < s) sm[tid] += sm[tid + s];
        __syncthreads();
    }
    if (tid == 0) dkl[0] = sm[0] / (float)MDIM;
}

extern "C" void kernel_launch(void* const* d_in, const int* in_sizes, int n_in,
                              void* d_out, int out_size, void* d_ws, size_t ws_size,
                              hipStream_t stream)
{
    const float* x      = (const float*)d_in[0];
    const float* E      = (const float*)d_in[1];
    const float* Mw     = (const float*)d_in[2];
    const float* W_xh   = (const float*)d_in[3];
    const float* b_xh   = (const float*)d_in[4];
    const float* W_hmu  = (const float*)d_in[5];
    const float* b_hmu  = (const float*)d_in[6];
    const float* W_hlvi = (const float*)d_in[7];
    const float* b_hlvi = (const float*)d_in[8];
    const float* W_hlvo = (const float*)d_in[9];
    const float* b_hlvo = (const float*)d_in[10];

    float* out = (float*)d_out;                  // [256*512] then [1] D_KL
    float* dkl = out + MDIM * ODIM;

    float* ws    = (float*)d_ws;
    float* h     = ws;                            // 256*256
    float* mu    = h     + MDIM * HDIM;           // 256*512
    float* sr    = mu    + MDIM * RDIM;
    float* sc    = sr    + MDIM * RDIM;
    float* lvr   = sc    + MDIM * ODIM;
    float* lvc   = lvr   + MDIM * RDIM;
    float* avec  = lvc   + MDIM * ODIM;
    float* bvec  = avec  + MDIM * RDIM;
    float* outmu = bvec  + MDIM * RDIM;
    float* SM    = outmu + MDIM * ODIM;           // 512
    float* klp   = SM    + RDIM;                  // 256*8
    float* xp    = klp   + MDIM * 8;              // 256*512 (padded x)
    float* Wp    = xp    + MDIM * 512;            // 512*256 (padded W_xh)

    k_pad        <<<dim3((MDIM * 512 + 512 * HDIM) / 256), 256, 0, stream>>>(
        x, W_xh, xp, Wp);
    k_hyper_h    <<<dim3(HDIM / 16, MDIM / 16),    32, 0, stream>>>(xp, Wp, b_xh, h);
    k_hyper_heads<<<dim3(RDIM / 16, MDIM / 16, 3), 32, 0, stream>>>(
        h, W_hmu, b_hmu, W_hlvi, b_hlvi, W_hlvo, b_hlvo, mu, sr, lvr, sc, lvc);
    k_msq        <<<dim3(RDIM / 256), 256, 0, stream>>>(Mw, SM);
    k_prep       <<<dim3(MDIM), 256, 0, stream>>>(x, mu, sr, lvr, sc, lvc, SM,
                                                  avec, bvec, klp);
    k_mu_out     <<<dim3(ODIM / 16, MDIM / 16), 32, 0, stream>>>(avec, Mw, outmu);
    k_estream    <<<dim3(MDIM), 256, 0, stream>>>(E, bvec, sc, outmu, out);
    k_kl         <<<dim3(1), 256, 0, stream>>>(klp, dkl);
}